// MultiHeadAttentionLayer_46660524703956
// MI455X (gfx1250) — compile-verified
//
#include <hip/hip_runtime.h>
#include <hip/hip_bf16.h>
#include <math.h>

// Problem constants (from reference)
#define BB   8
#define NN   256
#define IND  64
#define HH   8
#define DD   8
#define FF   64          // HH*DD
#define MROWS (BB*NN)    // 2048 rows of h

typedef __attribute__((ext_vector_type(2))) float v2f;
typedef __attribute__((ext_vector_type(4))) float v4f;
typedef __attribute__((ext_vector_type(8))) float v8f;

// ---------------------------------------------------------------------------
// Kernel 1: fused QKV projection using V_WMMA_F32_16X16X4_F32.
// One wave (32 threads) computes one 16x16 f32 output tile; K=64 in 16 steps
// of 4.  w selects {Wq, Wk, Wv}.  K is pre-scaled by 8^-0.5 and K/V are
// stored interleaved (float2 per channel) so the streaming kernel loads a
// lane's two channels with a single b128.
// ---------------------------------------------------------------------------
__global__ __launch_bounds__(32)
void mha_qkv_wmma(const float* __restrict__ h,
                  const float* __restrict__ Wq,
                  const float* __restrict__ Wk,
                  const float* __restrict__ Wv,
                  float* __restrict__ Q,    // [2048][64]
                  float* __restrict__ KV)   // [2048][64][2]  (.x = K*scale, .y = V)
{
    const int bid  = blockIdx.x;       // 0 .. 3*128*4 - 1
    const int w    = bid >> 9;         // / (128*4): 0=Q, 1=K, 2=V
    const int rem  = bid & 511;
    const int mt   = rem >> 2;         // 0..127  (M tile)
    const int nt   = rem & 3;          // 0..3    (N tile)
    const int lane = threadIdx.x;      // 0..31

    const float* W = (w == 0) ? Wq : ((w == 1) ? Wk : Wv);
    const float scale = (w == 1) ? 0.35355339059327373f : 1.0f;

    const int m0    = mt << 4;
    const int n0    = nt << 4;
    const int row   = m0 + (lane & 15);      // A row for this lane
    const int col   = n0 + (lane & 15);      // B/D column for this lane
    const int khalf = (lane >> 4) << 1;      // 0 for lanes 0-15, 2 for 16-31

    v8f c = {};
    #pragma unroll
    for (int k0 = 0; k0 < IND; k0 += 4) {
        // A: 16x4 f32 tile of h.  Lane l<16: {K0,K1}; lane>=16: {K2,K3}.
        v2f a;
        a.x = h[row * IND + k0 + khalf];
        a.y = h[row * IND + k0 + khalf + 1];
        // B: 4x16 f32 tile of W (row-major K x F).  Mirrored lane split.
        v2f bm;
        bm.x = W[(k0 + khalf)     * FF + col];
        bm.y = W[(k0 + khalf + 1) * FF + col];
        c = __builtin_amdgcn_wmma_f32_16x16x4_f32(
                /*neg_a=*/false, a, /*neg_b=*/false, bm,
                /*c_mod=*/(short)0, c, /*reuse_a=*/false, /*reuse_b=*/false);
    }

    // D layout: VGPR r -> M = r (lanes 0-15) / M = 8+r (lanes 16-31), N = col.
    const int mbase = m0 + ((lane >> 4) << 3);
    #pragma unroll
    for (int r = 0; r < 8; ++r) {
        const float d   = c[r] * scale;
        const int   idx = (mbase + r) * FF + col;
        if (w == 0)      Q[idx]          = d;   // Q
        else if (w == 1) KV[idx * 2]     = d;   // K * 8^-0.5
        else             KV[idx * 2 + 1] = d;   // V
    }
}

// ---------------------------------------------------------------------------
// Kernel 2: streaming attention with single-pass (online) softmax.
// One wave per (b,i) row; lane owns channels f0 = 2*lane, f0+1.
// Per j:
//   - KV b128 load  -> (K_f0, V_f0, K_f1, V_f1)      [L2-resident, RT]
//   - e_att/e_value b64 non-temporal loads            [single-use stream]
//   - per-head qk: lane partial covers d = {2*(lane%4), 2*(lane%4)+1} of the head;
//     head = 4 lanes -> 2-step shfl_xor butterfly.
//   - online-softmax update of (m, s, acc) for both channels.
// ---------------------------------------------------------------------------
__global__ __launch_bounds__(64)
void mha_stream_attn(const float* __restrict__ e_att,
                     const float* __restrict__ e_value,
                     const unsigned char* __restrict__ attn_mask, // numpy bool
                     const float* __restrict__ Q,
                     const float* __restrict__ KV,
                     float* __restrict__ out)
{
    const int lane = threadIdx.x & 31;
    const int bi   = (blockIdx.x << 1) + (threadIdx.x >> 5); // 0..2047
    const int b    = bi >> 8;
    const int f0   = lane << 1;                              // 0,2,..,62

    const v2f qf = *(const v2f*)(Q + bi * FF + f0);          // (Q_f0, Q_f1)

    const v2f* ea2  = (const v2f*)(e_att   + bi * NN * FF);  // [j][32 lanes]
    const v2f* ev2  = (const v2f*)(e_value + bi * NN * FF);
    const unsigned char* mrow = attn_mask + bi * NN;
    const v4f* kvb  = (const v4f*)KV + (b << 8) * (FF / 2);  // [j][32 lanes]

    float m0 = -3.0e38f, s0 = 0.0f, a0 = 0.0f;
    float m1 = -3.0e38f, s1 = 0.0f, a1 = 0.0f;

    #pragma unroll 4
    for (int j = 0; j < NN; ++j) {
        const v4f kv = kvb[j * 32 + lane];        // K_f0, V_f0, K_f1, V_f1

        // Per-head dot over d: this lane holds d = {f0%8, f0%8+1} of head f0/8.
        float prod = qf.x * kv[0] + qf.y * kv[2];
        prod += __shfl_xor(prod, 1, 32);
        prod += __shfl_xor(prod, 2, 32);          // qk for the 4-lane head group

        const v2f eav = __builtin_nontemporal_load(ea2 + j * 32 + lane);
        const v2f evv = __builtin_nontemporal_load(ev2 + j * 32 + lane);
        const float mk = (float)mrow[j];

        float l0 = prod + eav.x;
        float l1 = prod + eav.y;
        l0 = (mk != 0.0f) ? l0 : (l0 - 1.0e24f);
        l1 = (mk != 0.0f) ? l1 : (l1 - 1.0e24f);

        const float vt0 = kv[1] + evv.x;
        const float vt1 = kv[3] + evv.y;

        // Online softmax, channel 0
        {
            const float nm = fmaxf(m0, l0);
            const float sc = __expf(m0 - nm);
            const float p  = __expf(l0 - nm);
            s0 = s0 * sc + p;                 // denominator keeps masked terms
            a0 = a0 * sc + p * mk * vt0;      // numerator masked (scores*=mask)
            m0 = nm;
        }
        // Online softmax, channel 1
        {
            const float nm = fmaxf(m1, l1);
            const float sc = __expf(m1 - nm);
            const float p  = __expf(l1 - nm);
            s1 = s1 * sc + p;
            a1 = a1 * sc + p * mk * vt1;
            m1 = nm;
        }
    }

    v2f o;
    o.x = a0 / s0;
    o.y = a1 / s1;
    *(v2f*)(out + bi * FF + f0) = o;
}

// ---------------------------------------------------------------------------
extern "C" void kernel_launch(void* const* d_in, const int* in_sizes, int n_in,
                              void* d_out, int out_size, void* d_ws, size_t ws_size,
                              hipStream_t stream)
{
    const float*         h       = (const float*)d_in[0];
    const float*         e_att   = (const float*)d_in[1];
    const float*         e_value = (const float*)d_in[2];
    const unsigned char* mask    = (const unsigned char*)d_in[3];
    const float*         Wq      = (const float*)d_in[4];
    const float*         Wk      = (const float*)d_in[5];
    const float*         Wv      = (const float*)d_in[6];
    float*               out     = (float*)d_out;

    // Workspace layout: Q [2048*64] f32, then KV interleaved [2048*64] float2.
    float* Q  = (float*)d_ws;
    float* KV = Q + MROWS * FF;          // 512 KiB after Q; total 1.5 MiB

    // Kernel 1: 3 weights * 128 M-tiles * 4 N-tiles waves.
    mha_qkv_wmma<<<3 * 128 * 4, 32, 0, stream>>>(h, Wq, Wk, Wv, Q, KV);

    // Kernel 2: one wave per (b, i) row, two rows per block.
    mha_stream_attn<<<MROWS / 2, 64, 0, stream>>>(e_att, e_value, mask, Q, KV, out);
}